// HmmTrajNet2_52621939310677
// MI455X (gfx1250) — compile-verified
//
#include <hip/hip_runtime.h>
#include <cstdint>

#define DEVINL __device__ __forceinline__

constexpr int B_   = 64;    // batch
constexpr int T_   = 512;   // time steps
constexpr int TP1  = 513;
constexpr int NB   = 64;    // hidden states b
constexpr int NA   = 18;    // actions
constexpr int CH   = 8;     // scan steps per async chunk
constexpr int NFULL = 63;   // 63 full chunks of 8 + 1 epilogue chunk of 7 = 511 steps
constexpr int TAIL  = (T_ - 1) - NFULL * CH;   // 7
constexpr float NEG_INF = -1.0e30f;            // finite sentinel: exp() underflows to 0

// ---- CDNA5 async global->LDS copies (ASYNCcnt-tracked DMA path) -------------
// LDS destination address = low 32 bits of the generic pointer (flat LDS
// aperture keeps the DS-relative offset in addr[31:0]).
DEVINL unsigned lds_addr(const void* p) { return (unsigned)(uintptr_t)p; }

DEVINL void async_b32(unsigned lds, const void* g) {
  asm volatile("global_load_async_to_lds_b32 %0, %1, off" :: "v"(lds), "v"(g) : "memory");
}
DEVINL void async_b64(unsigned lds, const void* g) {
  asm volatile("global_load_async_to_lds_b64 %0, %1, off" :: "v"(lds), "v"(g) : "memory");
}
DEVINL void async_b128(unsigned lds, const void* g) {
  asm volatile("global_load_async_to_lds_b128 %0, %1, off" :: "v"(lds), "v"(g) : "memory");
}

template <int N> DEVINL void wait_asynccnt() {
#if __has_builtin(__builtin_amdgcn_s_wait_asynccnt)
  __builtin_amdgcn_s_wait_asynccnt(N);
#else
  asm volatile("s_wait_asynccnt %0" :: "n"(N) : "memory");
#endif
}

// ---- wave32 butterfly reductions (no barriers needed: 1 wave per WG) --------
DEVINL float wave_max32(float x) {
#pragma unroll
  for (int off = 16; off > 0; off >>= 1) x = fmaxf(x, __shfl_xor(x, off, 32));
  return x;
}
DEVINL float wave_sum32(float x) {
#pragma unroll
  for (int off = 16; off > 0; off >>= 1) x += __shfl_xor(x, off, 32);
  return x;
}
DEVINL float logaddexp_f(float a, float b) {
  float mx = fmaxf(a, b), mn = fminf(a, b);
  return mx + __logf(1.0f + __expf(mn - mx));
}

// ---- async-issue one chunk of STEPS scan rows into parity buffer ------------
template <int STEPS>
DEVINL void issue_chunk(int t0, int lane,
                        const float* stopBase, const float* startBase, const float* actBase,
                        const int* ldsActs,
                        float* stopB, float* startB, float* chosenB)
{
#pragma unroll
  for (int r = 0; r < STEPS; ++r) {
    const int t = t0 + r;
    const int a = ldsActs[t];   // wave-uniform LDS read
    // stop_logps row: 4 contiguous floats per lane (beta0,ombeta0,beta1,ombeta1)
    async_b128(lds_addr(&stopB[(r * NB + lane * 2) * 2]),
               stopBase + ((size_t)t * NB + lane * 2) * 2);
    // start row: 2 floats per lane
    async_b64(lds_addr(&startB[r * NB + lane * 2]),
              startBase + (size_t)t * NB + lane * 2);
    // chosen = per-lane GATHER from action_logps (per-lane VGPR addresses)
    const float* cp = actBase + ((size_t)t * NB + lane * 2) * NA + a;
    async_b32(lds_addr(&chosenB[r * NB + lane * 2]),     cp);
    async_b32(lds_addr(&chosenB[r * NB + lane * 2 + 1]), cp + NA);
  }
}

// ---- run STEPS scan steps out of the landed parity buffer -------------------
template <int STEPS>
DEVINL void compute_chunk(int t0, int lane, int tcap,
                          const float* stopB, const float* startB, const float* chosenB,
                          float& f0, float& f1, float& g0, float& g1)
{
#pragma unroll
  for (int r = 0; r < STEPS; ++r) {
    const int t = t0 + r;
    const float4 sp = *reinterpret_cast<const float4*>(&stopB[(r * NB + lane * 2) * 2]);
    const float2 st = *reinterpret_cast<const float2*>(&startB[r * NB + lane * 2]);
    const float2 ch = *reinterpret_cast<const float2*>(&chosenB[r * NB + lane * 2]);
    // logS = logsumexp_j(f_prev[j] + beta_t[j]) over all 64 states (one wave)
    const float v0 = f0 + sp.x;
    const float v1 = f1 + sp.z;
    const float m  = wave_max32(fmaxf(v0, v1));
    const float s  = wave_sum32(__expf(v0 - m) + __expf(v1 - m));
    const float logS = m + __logf(s);
    // f_new[k] = chosen[k] + logaddexp(start[k] + logS, f_prev[k] + ombeta[k])
    f0 = ch.x + logaddexp_f(st.x + logS, f0 + sp.y);
    f1 = ch.y + logaddexp_f(st.y + logS, f1 + sp.w);
    if (t == tcap) { g0 = f0; g1 = f1; }
  }
}

// ---- main scan: one wave32 per batch element --------------------------------
__global__ __launch_bounds__(32)
void hmm_scan_kernel(const float* __restrict__ stop_logps,    // (B,513,64,2)
                     const float* __restrict__ start_logps,   // (B,513,64)
                     const float* __restrict__ action_logps,  // (B,513,64,18)
                     const int*   __restrict__ actions,       // (B,512)
                     const int*   __restrict__ lengths,       // (B,)
                     float* __restrict__ batch_out)           // (B,) total_logps
{
  __shared__ __align__(16) float ldsStop[2][CH * 2 * NB];   // (beta,ombeta) pairs
  __shared__ __align__(16) float ldsStart[2][CH * NB];
  __shared__ __align__(16) float ldsChosen[2][CH * NB];
  __shared__ int ldsActs[T_];

  const int lane = threadIdx.x;       // lane L owns states 2L and 2L+1
  const int bi   = blockIdx.x;

  const float* stopBase  = stop_logps   + (size_t)bi * TP1 * NB * 2;
  const float* startBase = start_logps  + (size_t)bi * TP1 * NB;
  const float* actBase   = action_logps + (size_t)bi * TP1 * NB * NA;

  // Stage action ids once (compile-time trip count: no branchy EXEC games).
#pragma unroll
  for (int j = 0; j < T_ / 32; ++j)
    ldsActs[j * 32 + lane] = actions[(size_t)bi * T_ + j * 32 + lane];
  __syncthreads();   // single-wave WG: S_NOP + compiler ordering fence

  const int len  = lengths[bi];
  const int tcap = len - 1;                   // f_all index to capture (0..511)

  float f0 = (lane == 0) ? 0.0f : NEG_INF;    // state 2*lane
  float f1 = NEG_INF;                         // state 2*lane+1
  float g0 = f0, g1 = f1;                     // covers tcap == 0

  // Software pipeline: 63 full 8-step chunks + one 7-step epilogue chunk,
  // double-buffered through LDS via the async DMA path.
  issue_chunk<CH>(1, lane, stopBase, startBase, actBase, ldsActs,
                  ldsStop[0], ldsStart[0], ldsChosen[0]);

  for (int c = 0; c < NFULL; ++c) {
    const int p = c & 1;
    const int q = p ^ 1;
    if (c < NFULL - 1) {
      issue_chunk<CH>(1 + (c + 1) * CH, lane, stopBase, startBase, actBase, ldsActs,
                      ldsStop[q], ldsStart[q], ldsChosen[q]);
    } else {
      issue_chunk<TAIL>(1 + NFULL * CH, lane, stopBase, startBase, actBase, ldsActs,
                        ldsStop[q], ldsStart[q], ldsChosen[q]);
    }
    wait_asynccnt<4 * (CH - 1)>();            // <=28 outstanding => chunk c landed
    compute_chunk<CH>(1 + c * CH, lane, tcap,
                      ldsStop[p], ldsStart[p], ldsChosen[p], f0, f1, g0, g1);
  }
  wait_asynccnt<0>();                          // epilogue chunk landed
  compute_chunk<TAIL>(1 + NFULL * CH, lane, tcap,
                      ldsStop[NFULL & 1], ldsStart[NFULL & 1], ldsChosen[NFULL & 1],
                      f0, f1, g0, g1);

  // total_logps[bi] = logsumexp_k( f_all[tcap][k] + stop_logps[bi, len, k, STOP] )
  const float4 x1 = *reinterpret_cast<const float4*>(
      stopBase + ((size_t)len * NB + lane * 2) * 2);
  const float w0 = g0 + x1.x;                 // STOP component of state 2*lane
  const float w1 = g1 + x1.z;                 // STOP component of state 2*lane+1
  const float mm = wave_max32(fmaxf(w0, w1));
  const float ss = wave_sum32(__expf(w0 - mm) + __expf(w1 - mm));
  if (lane == 0) batch_out[bi] = mm + __logf(ss);
}

// ---- final reduction: -sum over 64 batch totals -----------------------------
__global__ __launch_bounds__(32)
void hmm_finalize_kernel(const float* __restrict__ batch_out, float* __restrict__ out)
{
  const int lane = threadIdx.x;
  float s = batch_out[lane] + batch_out[lane + 32];
  s = wave_sum32(s);
  if (lane == 0) out[0] = -s;
}

extern "C" void kernel_launch(void* const* d_in, const int* in_sizes, int n_in,
                              void* d_out, int out_size, void* d_ws, size_t ws_size,
                              hipStream_t stream) {
  (void)in_sizes; (void)n_in; (void)out_size; (void)ws_size;
  const float* stop    = (const float*)d_in[0];
  const float* start   = (const float*)d_in[1];
  const float* act     = (const float*)d_in[2];
  const int*   actions = (const int*)d_in[3];
  const int*   lengths = (const int*)d_in[4];
  float* ws = (float*)d_ws;   // 64 floats of scratch

  hmm_scan_kernel<<<B_, 32, 0, stream>>>(stop, start, act, actions, lengths, ws);
  hmm_finalize_kernel<<<1, 32, 0, stream>>>(ws, (float*)d_out);
}